// NodeModel_20358144983598
// MI455X (gfx1250) — compile-verified
//
#include <hip/hip_runtime.h>

#define H 64
#define TM 16   // nodes per WMMA tile (M)
#define NW 4    // waves per block in MLP kernel

typedef __attribute__((ext_vector_type(2))) float v2f;
typedef __attribute__((ext_vector_type(8))) float v8f;

// ---------------------------------------------------------------- zero scratch
__global__ void zero_kernel(float* __restrict__ p, int n) {
    int i = blockIdx.x * blockDim.x + threadIdx.x;
    int stride = gridDim.x * blockDim.x;
    for (; i < n; i += stride) p[i] = 0.0f;
}

// ------------------------------------------------------- scatter-sum + counts
// 64 consecutive threads handle one edge (coalesced 256B read of e[edge,:]),
// fp32 atomics accumulate into sums[receiver,:] (12.8MB, L2-resident).
__global__ void scatter_kernel(const float* __restrict__ e,
                               const long long* __restrict__ edge_index,
                               float* __restrict__ sums,
                               float* __restrict__ counts,
                               int n_edges) {
    long long tid = (long long)blockIdx.x * blockDim.x + threadIdx.x;
    int edge = (int)(tid >> 6);
    int f    = (int)(tid & 63);
    if (edge >= n_edges) return;
    int r = (int)edge_index[(long long)n_edges + edge];   // receiver = row 1
    atomicAdd(&sums[(long long)r * H + f], e[(long long)edge * H + f]);
    if (f == 0) atomicAdd(&counts[r], 1.0f);
}

// ---------------------------------------------------- pack weights pair-wise
// Global W is K x 64 row-major. LDS layout pairs consecutive K rows so a B
// fragment (W[kk][n], W[kk+1][n]) is one aligned ds_load_b64:
//   sWp[(k>>1)*128 + 2*n + (k&1)] = W[k][n]
__device__ __attribute__((always_inline))
void stage_packed(const float* __restrict__ G, float* sWp, int K, int tid, int bd) {
    for (int i = tid; i < K * H; i += bd) {
        int k = i >> 6;          // K row
        int n = i & 63;          // output column
        sWp[(k >> 1) * 128 + 2 * n + (k & 1)] = G[i];
    }
}

// ------------------------------------------------- one dense layer via WMMA f32
// sIn: 16 x K activation tile in LDS (row stride ldi floats)
// sWp: packed K x 64 weights in LDS (see stage_packed), bias: 64 floats in LDS
// sOut: 16 x 64 tile in LDS (row stride H)
// A layout (16xK f32): lane holds row = lane%16, K-pair {4k+2*(lane/16), +1}
// B layout (4x16 f32): lane holds col = lane%16, same K-pair selection
// C/D layout (16x16 f32): vgpr r of lane = D[r + 8*(lane/16)][lane%16]
__device__ __attribute__((always_inline))
void wmma_layer(const float* sIn, int ldi, const float* sWp, const float* bias,
                float* sOut, int K, int col, int half, bool do_relu) {
    v8f acc0 = {}, acc1 = {}, acc2 = {}, acc3 = {};
    for (int k = 0; k < K / 4; ++k) {
        int kk = 4 * k + 2 * half;                       // even K index for lane
        v2f a = *(const v2f*)&sIn[col * ldi + kk];       // ds_load_b64
        const float* bp = &sWp[(kk >> 1) * 128 + 2 * col];
        v2f b;
        b = *(const v2f*)&bp[0];                         // n-tile 0
        acc0 = __builtin_amdgcn_wmma_f32_16x16x4_f32(false, a, false, b, (short)0, acc0, false, false);
        b = *(const v2f*)&bp[32];                        // n-tile 1
        acc1 = __builtin_amdgcn_wmma_f32_16x16x4_f32(false, a, false, b, (short)0, acc1, false, false);
        b = *(const v2f*)&bp[64];                        // n-tile 2
        acc2 = __builtin_amdgcn_wmma_f32_16x16x4_f32(false, a, false, b, (short)0, acc2, false, false);
        b = *(const v2f*)&bp[96];                        // n-tile 3
        acc3 = __builtin_amdgcn_wmma_f32_16x16x4_f32(false, a, false, b, (short)0, acc3, false, false);
    }
    float bi0 = bias[col], bi1 = bias[16 + col], bi2 = bias[32 + col], bi3 = bias[48 + col];
#pragma unroll
    for (int r = 0; r < 8; ++r) {
        int row = r + 8 * half;
        float y0 = acc0[r] + bi0, y1 = acc1[r] + bi1, y2 = acc2[r] + bi2, y3 = acc3[r] + bi3;
        if (do_relu) {
            y0 = fmaxf(y0, 0.0f); y1 = fmaxf(y1, 0.0f);
            y2 = fmaxf(y2, 0.0f); y3 = fmaxf(y3, 0.0f);
        }
        sOut[row * H +  0 + col] = y0;
        sOut[row * H + 16 + col] = y1;
        sOut[row * H + 32 + col] = y2;
        sOut[row * H + 48 + col] = y3;
    }
}

// ----------------------------------------------------------------- MLP kernel
// Per wave: 16-node tile. x = [sums/max(cnt,1) | v] (16x128) -> 3 WMMA layers.
__global__ void mlp_kernel(const float* __restrict__ v,
                           const float* __restrict__ sums,
                           const float* __restrict__ counts,
                           const float* __restrict__ W0, const float* __restrict__ b0,
                           const float* __restrict__ W1, const float* __restrict__ b1,
                           const float* __restrict__ W2, const float* __restrict__ b2,
                           float* __restrict__ out, int n_nodes) {
    extern __shared__ float smem[];
    float* sW0 = smem;                 // 2*H*H = 8192 (packed)
    float* sW1 = sW0 + 2 * H * H;      // H*H  = 4096 (packed)
    float* sW2 = sW1 + H * H;          // H*H  = 4096 (packed)
    float* sB  = sW2 + H * H;          // 3*H  = 192
    float* sX  = sB + 3 * H;           // NW * 16 * 128
    float* sY  = sX + NW * TM * 2 * H; // NW * 16 * 64

    const int tid = threadIdx.x;
    const int bd  = blockDim.x;

    stage_packed(W0, sW0, 2 * H, tid, bd);
    stage_packed(W1, sW1, H,     tid, bd);
    stage_packed(W2, sW2, H,     tid, bd);
    for (int i = tid; i < H; i += bd) {
        sB[i] = b0[i]; sB[H + i] = b1[i]; sB[2 * H + i] = b2[i];
    }

    const int wave = tid >> 5;
    const int lane = tid & 31;
    const int col  = lane & 15;
    const int half = lane >> 4;

    const int n_tiles = (n_nodes + TM - 1) / TM;
    int tile = blockIdx.x * NW + wave;
    const bool valid = (tile < n_tiles);
    if (!valid) tile = n_tiles - 1;     // clamp: keep EXEC uniform for WMMA

    float* sXw = sX + wave * (TM * 2 * H);
    float* sYw = sY + wave * (TM * H);

    // Stage x tile = [mean | v] into LDS (coalesced float4 global loads)
    for (int idx = lane; idx < TM * 32; idx += 32) {
        int row = idx >> 5;       // node row 0..15
        int q   = idx & 31;       // float4 slot in 128-float row
        long long node = (long long)tile * TM + row;
        if (node >= n_nodes) node = n_nodes - 1;   // clamp (never stored)
        float4 f4;
        if (q < 16) {
            f4 = ((const float4*)sums)[node * (H / 4) + q];
            float rc = 1.0f / fmaxf(counts[node], 1.0f);
            f4.x *= rc; f4.y *= rc; f4.z *= rc; f4.w *= rc;
        } else {
            f4 = ((const float4*)v)[node * (H / 4) + (q - 16)];
        }
        ((float4*)sXw)[row * 32 + q] = f4;
    }
    __syncthreads();

    // layer 0: (16x128)@(128x64)+b0, relu  -> sYw
    wmma_layer(sXw, 2 * H, sW0, sB,         sYw, 2 * H, col, half, true);
    __syncthreads();
    // layer 1: (16x64)@(64x64)+b1, relu    -> sXw (reused, stride H)
    wmma_layer(sYw, H,     sW1, sB + H,     sXw, H,     col, half, true);
    __syncthreads();
    // layer 2: (16x64)@(64x64)+b2, no relu -> sYw
    wmma_layer(sXw, H,     sW2, sB + 2 * H, sYw, H,     col, half, false);
    __syncthreads();

    // coalesced float4 store of the 16x64 result
    for (int idx = lane; idx < TM * 16; idx += 32) {
        int row = idx >> 4;
        int q   = idx & 15;
        long long node = (long long)tile * TM + row;
        if (valid && node < n_nodes)
            ((float4*)out)[node * (H / 4) + q] = ((const float4*)sYw)[row * 16 + q];
    }
}

// ---------------------------------------------------------------------- launch
extern "C" void kernel_launch(void* const* d_in, const int* in_sizes, int n_in,
                              void* d_out, int out_size, void* d_ws, size_t ws_size,
                              hipStream_t stream) {
    const float*     v  = (const float*)d_in[0];
    const long long* ei = (const long long*)d_in[1];
    const float*     e  = (const float*)d_in[2];
    const float*     W0 = (const float*)d_in[3];
    const float*     b0 = (const float*)d_in[4];
    const float*     W1 = (const float*)d_in[5];
    const float*     b1 = (const float*)d_in[6];
    const float*     W2 = (const float*)d_in[7];
    const float*     b2 = (const float*)d_in[8];
    float* out = (float*)d_out;

    const int n_nodes = in_sizes[0] / H;
    const int n_edges = in_sizes[2] / H;

    float* sums   = (float*)d_ws;                       // n_nodes * 64
    float* counts = sums + (size_t)n_nodes * H;         // n_nodes

    // 1) zero accumulators (sums + counts are contiguous)
    int nz = n_nodes * (H + 1);
    zero_kernel<<<(nz + 255) / 256, 256, 0, stream>>>(sums, nz);

    // 2) atomic scatter-sum of edge features + edge counts
    long long total = (long long)n_edges * H;
    int sblocks = (int)((total + 255) / 256);
    scatter_kernel<<<sblocks, 256, 0, stream>>>(e, ei, sums, counts, n_edges);

    // 3) WMMA MLP over 16-node tiles
    const int n_tiles = (n_nodes + TM - 1) / TM;
    const int grid = (n_tiles + NW - 1) / NW;
    const size_t smem_bytes =
        (size_t)(2 * H * H + H * H + H * H + 3 * H + NW * TM * 2 * H + NW * TM * H)
        * sizeof(float);   // ~113 KB, legal on CDNA5 (320 KB LDS / WGP)
    hipFuncSetAttribute(reinterpret_cast<const void*>(mlp_kernel),
                        hipFuncAttributeMaxDynamicSharedMemorySize, (int)smem_bytes);
    mlp_kernel<<<grid, NW * 32, smem_bytes, stream>>>(v, sums, counts,
                                                      W0, b0, W1, b1, W2, b2,
                                                      out, n_nodes);
}